// PillarLayer_29583734735022
// MI455X (gfx1250) — compile-verified
//
#include <hip/hip_runtime.h>
#include <stdint.h>

// ---------------- problem constants (from reference) ----------------
static constexpr int BS    = 4;
static constexpr int P_PER = 12000;
static constexpr int XL    = 432;
static constexpr int YL    = 496;
static constexpr int NPTS  = 32;
static constexpr long long PLANE = (long long)YL * XL;          // 214272

// flat offsets (in floats) of each tuple member inside d_out
static constexpr long long O_PILLARS = 0;                                    // 4,608,000
static constexpr long long O_COORS   = (long long)BS * P_PER * NPTS * 3;     // 4,608,000
static constexpr long long O_NPTS    = O_COORS + (long long)BS * P_PER * 3;  // 4,752,000
static constexpr long long O_ALL     = O_NPTS  + (long long)BS * P_PER;      // 4,800,000
static constexpr long long SZ_ALL    = (long long)BS * 3 * NPTS * PLANE;     // 82,280,448
static constexpr long long O_CENTER  = O_ALL + SZ_ALL;                       // 87,080,448
static constexpr long long SZ_CENTER = (long long)BS * 3 * PLANE;            // 2,571,264

typedef float v2f __attribute__((ext_vector_type(2)));
typedef float v4f __attribute__((ext_vector_type(4)));
typedef float v8f __attribute__((ext_vector_type(8)));

// ---------------------------------------------------------------
// Kernel 1: zero-fill pillar_all + pillar_center with NT b128 stores.
// Region (339 MB) is larger than the 192 MB L2 -> non-temporal.
// ---------------------------------------------------------------
__global__ void pillar_fill_kernel(float* __restrict__ out, long long n4) {
    v4f z = {0.f, 0.f, 0.f, 0.f};
    v4f* dst = (v4f*)(out + O_ALL);               // O_ALL*4 bytes is 16B-aligned
    long long i = (long long)blockIdx.x * blockDim.x + threadIdx.x;
    long long stride = (long long)gridDim.x * blockDim.x;
    for (; i < n4; i += stride)
        __builtin_nontemporal_store(z, dst + i);
}

// ---------------------------------------------------------------
// Kernel 2: one wave32 per pillar. Lane n owns point n.
//  - coalesced 384B read + 384B write for the pillars passthrough
//  - 3 scattered dword stores per lane into pillar_all (random iy,ix)
//  - lanes 0..2 copy coors (as float), lane 3 copies npoints (as float)
// ---------------------------------------------------------------
__global__ void pillar_scatter_kernel(const float* __restrict__ pillars,
                                      const int*   __restrict__ coors,
                                      const int*   __restrict__ npoints,
                                      float* __restrict__ out, int P) {
    const int w    = (int)((blockIdx.x * blockDim.x + threadIdx.x) >> 5);
    const int lane = threadIdx.x & 31;
    if (w >= P) return;                    // wave-uniform
    const int p = w;

    const int b  = coors[p * 3 + 0];
    const int ix = coors[p * 3 + 1];
    const int iy = coors[p * 3 + 2];

    const float* src = pillars + (size_t)p * 96 + lane * 3;
    const float x = src[0], y = src[1], z = src[2];

    // passthrough copy (coalesced: wave writes 384 contiguous bytes)
    float* o1 = out + O_PILLARS + (size_t)p * 96 + lane * 3;
    o1[0] = x; o1[1] = y; o1[2] = z;

    if (lane < 3)   out[O_COORS + (size_t)p * 3 + lane] = (float)coors[p * 3 + lane];
    if (lane == 3)  out[O_NPTS  + p] = (float)npoints[p];

    // pillar_all[b][c][n][iy][ix], n = lane
    const long long base = O_ALL + (long long)b * 3 * NPTS * PLANE
                         + (long long)iy * XL + ix;
    out[base + (long long)(0 * NPTS + lane) * PLANE] = x;
    out[base + (long long)(1 * NPTS + lane) * PLANE] = y;
    out[base + (long long)(2 * NPTS + lane) * PLANE] = z;
}

// ---------------------------------------------------------------
// Kernel 3: centers via WMMA. One wave handles 16 pillars.
//  - stage 16 pillars (6144 B) into LDS with global_load_async_to_lds_b128
//  - sum over the 32 points with V_WMMA_F32_16X16X4_F32, B = ones:
//      D[m][n] = sum_k A[m][k] * 1  (accumulated over 8 chunks of K=4,
//      one accumulator per channel)
//  - divide by npoints, scatter into pillar_center[b][c][iy][ix]
// LDS layout (dynamic shared only, so async offsets start at 0):
//   stage: wpb * 1536 floats   (wave w at w*1536)
//   sums : wpb * 48   floats
// ---------------------------------------------------------------
__global__ void pillar_center_wmma_kernel(const float* __restrict__ pillars,
                                          const int*   __restrict__ coors,
                                          const int*   __restrict__ npoints,
                                          float* __restrict__ out, int P) {
    extern __shared__ float smem[];
    const int lane = threadIdx.x & 31;
    const int w    = threadIdx.x >> 5;
    const int wpb  = blockDim.x >> 5;
    const int gw   = blockIdx.x * wpb + w;
    const int p0   = gw * 16;
    if (p0 >= P) return;   // grid sized exactly (P % 16 == 0) -> never taken

    float* stage = smem + (size_t)w * 1536;
    float* sums  = smem + (size_t)wpb * 1536 + (size_t)w * 48;

    // ---- async stage: 16 pillars * 384B = 6144B, 12 x (32 lanes x 16B) ----
    #pragma unroll
    for (int j = 0; j < 12; ++j) {
        unsigned ldsOff = (unsigned)((w * 1536 + j * 128 + lane * 4) * 4);
        uint64_t gaddr  = (uint64_t)(pillars + (size_t)p0 * 96 + j * 128 + lane * 4);
        asm volatile("global_load_async_to_lds_b128 %0, %1, off"
                     :: "v"(ldsOff), "v"(gaddr) : "memory");
    }
    asm volatile("s_wait_asynccnt 0x0" ::: "memory");

    // ---- WMMA reduction: A(16x4 f32) x ones(4x16) + C ----
    // A layout (ISA 7.12.2, 32-bit A 16x4): lane<16 -> K=k0,k0+1 ; lane>=16 -> K=k0+2,k0+3
    const int pl  = lane & 15;
    const int khi = (lane < 16) ? 0 : 2;
    const v2f ones = {1.0f, 1.0f};
    v8f accx = {}; v8f accy = {}; v8f accz = {};
    #pragma unroll
    for (int k0 = 0; k0 < NPTS; k0 += 4) {
        const float* r0 = stage + pl * 96 + (k0 + khi) * 3;      // point k, channels 0..2
        const float* r1 = stage + pl * 96 + (k0 + khi + 1) * 3;  // point k+1
        v2f a;
        a.x = r0[0]; a.y = r1[0];
        accx = __builtin_amdgcn_wmma_f32_16x16x4_f32(false, a, false, ones,
                                                     (short)0, accx, false, false);
        a.x = r0[1]; a.y = r1[1];
        accy = __builtin_amdgcn_wmma_f32_16x16x4_f32(false, a, false, ones,
                                                     (short)0, accy, false, false);
        a.x = r0[2]; a.y = r1[2];
        accz = __builtin_amdgcn_wmma_f32_16x16x4_f32(false, a, false, ones,
                                                     (short)0, accz, false, false);
    }

    // ---- extract D column N=0 (lanes 0 and 16 hold rows 0..7 / 8..15) ----
    if ((lane & 15) == 0) {
        const int mbase = (lane == 0) ? 0 : 8;
        #pragma unroll
        for (int r = 0; r < 8; ++r) {
            sums[0 * 16 + mbase + r] = accx[r];
            sums[1 * 16 + mbase + r] = accy[r];
            sums[2 * 16 + mbase + r] = accz[r];
        }
    }
    __syncthreads();

    if (lane < 16) {
        const int p  = p0 + lane;
        const float inv = 1.0f / (float)npoints[p];
        const float cx = sums[0 * 16 + lane] * inv;
        const float cy = sums[1 * 16 + lane] * inv;
        const float cz = sums[2 * 16 + lane] * inv;
        const int b  = coors[p * 3 + 0];
        const int ix = coors[p * 3 + 1];
        const int iy = coors[p * 3 + 2];
        const long long base = O_CENTER + (long long)b * 3 * PLANE
                             + (long long)iy * XL + ix;
        out[base + 0 * PLANE] = cx;
        out[base + 1 * PLANE] = cy;
        out[base + 2 * PLANE] = cz;
    }
}

// ---------------------------------------------------------------
extern "C" void kernel_launch(void* const* d_in, const int* in_sizes, int n_in,
                              void* d_out, int out_size, void* d_ws, size_t ws_size,
                              hipStream_t stream) {
    const float* pillars = (const float*)d_in[0];
    const int*   coors   = (const int*)d_in[1];
    const int*   npoints = (const int*)d_in[2];
    float*       out     = (float*)d_out;

    const int P = in_sizes[0] / (NPTS * 3);          // 48000

    // 1) zero-fill pillar_all + pillar_center (b128 NT stores)
    const long long n4 = (SZ_ALL + SZ_CENTER) / 4;   // 21,212,928 float4s
    pillar_fill_kernel<<<8192, 256, 0, stream>>>(out, n4);

    // 2) wave-per-pillar passthrough + scatter (P*32 threads)
    const int threads = 256;
    const int blocks2 = (P * 32 + threads - 1) / threads;   // 6000
    pillar_scatter_kernel<<<blocks2, threads, 0, stream>>>(pillars, coors, npoints, out, P);

    // 3) centers via async-LDS staging + WMMA (16 pillars / wave, 8 waves / block)
    const int wpb     = 8;                            // waves per block (256 threads)
    const int blocks3 = (P / 16 + wpb - 1) / wpb;     // 375
    const size_t shmem = (size_t)wpb * (1536 + 48) * sizeof(float);  // 50,688 B
    pillar_center_wmma_kernel<<<blocks3, wpb * 32, shmem, stream>>>(pillars, coors, npoints, out, P);
}